// Mismatch_loss_19018115187339
// MI455X (gfx1250) — compile-verified
//
#include <hip/hip_runtime.h>

typedef __attribute__((ext_vector_type(2))) float v2f;
typedef __attribute__((ext_vector_type(4))) float v4f;
typedef __attribute__((ext_vector_type(8))) float v8f;

#define HW        147456            // 384*384
#define SLICES    128               // B*C = 8*16
#define SPLIT     8                 // blocks per (b,c) slice
#define CHUNK     (HW / SPLIT)      // 18432 elements per block
#define THREADS   256               // 8 waves of 32
#define VEC_ITERS (CHUNK / (THREADS * 4))  // 18 float4 iters per thread

#define ALPHA  0.05f
#define SMOOTH 1e-6f

// ---------------------------------------------------------------------------
// Kernel A: streaming partial reduction. One block handles 18432 contiguous
// elements of one (b,c) slice. Emits 5 partials per block into d_ws:
// [slice][part][{sum_t, sum_d2, sum_d2t, max_t, max_m}]
// ---------------------------------------------------------------------------
__global__ __launch_bounds__(THREADS)
void mismatch_partial_kernel(const float* __restrict__ net_out,
                             const float* __restrict__ target,
                             const float* __restrict__ maxpos,
                             float* __restrict__ partials) {
  const int slice = blockIdx.x >> 3;            // / SPLIT
  const int part  = blockIdx.x & (SPLIT - 1);
  const size_t base = (size_t)slice * HW + (size_t)part * CHUNK;
  const int tid = threadIdx.x;

  const v4f* __restrict__ pn = (const v4f*)(net_out + base);
  const v4f* __restrict__ pt = (const v4f*)(target + base);
  const v4f* __restrict__ pm = (const v4f*)(maxpos + base);

  float s_t = 0.0f, s_d2 = 0.0f, s_d2t = 0.0f;
  float mx_t = -__builtin_inff();
  float mx_m = -__builtin_inff();

#pragma unroll
  for (int i = 0; i < VEC_ITERS; ++i) {
    const int idx = tid + i * THREADS;
    // Streamed once; keep it out of the caches (NT temporal hint).
    v4f t = __builtin_nontemporal_load(pt + idx);
    v4f n = __builtin_nontemporal_load(pn + idx);
    v4f m = __builtin_nontemporal_load(pm + idx);
    if (i + 2 < VEC_ITERS) {  // clamp: keep prefetch addresses in-bounds
      __builtin_prefetch((const void*)(pt + idx + 2 * THREADS), 0, 0);
      __builtin_prefetch((const void*)(pn + idx + 2 * THREADS), 0, 0);
      __builtin_prefetch((const void*)(pm + idx + 2 * THREADS), 0, 0);
    }
#pragma unroll
    for (int k = 0; k < 4; ++k) {
      const float tv = t[k];
      const float dv = tv - n[k];
      const float d2 = dv * dv;
      s_t  += tv;
      s_d2 += d2;
      s_d2t = fmaf(d2, tv, s_d2t);
      mx_t = fmaxf(mx_t, tv);
      mx_m = fmaxf(mx_m, m[k]);
    }
  }

  // ---- block reduction: 256 threads -> 5 scalars ----
  __shared__ float red[3][THREADS];
  __shared__ float redm[2][THREADS];
  red[0][tid] = s_t;
  red[1][tid] = s_d2;
  red[2][tid] = s_d2t;
  redm[0][tid] = mx_t;
  redm[1][tid] = mx_m;
  __syncthreads();

  if (tid < 32) {  // wave 0 only: EXEC is all-1s here (WMMA requirement)
    // B = all-ones 4x16 matrix (layout-independent since every entry is 1.0).
    v2f bones; bones[0] = 1.0f; bones[1] = 1.0f;

    float out3[3];
#pragma unroll
    for (int q = 0; q < 3; ++q) {
      float v = 0.0f;
#pragma unroll
      for (int i = 0; i < THREADS / 32; ++i) v += red[q][tid + 32 * i];
      // A 16x4 f32 layout: VGPR0 lane l<16 -> (M=l,K=0); lane l>=16 -> (M=l-16,K=2).
      // With A-VGPR1 = 0:  D[m][n] = v[m] + v[m+16] for every n.
      v2f a; a[0] = v; a[1] = 0.0f;
      v8f c = {};
      v8f d = __builtin_amdgcn_wmma_f32_16x16x4_f32(
          /*neg_a=*/false, a, /*neg_b=*/false, bones,
          /*c_mod=*/(short)0, c, /*reuse_a=*/false, /*reuse_b=*/false);
      // Lane l<16 holds D rows 0..7 (one column); lane l>=16 holds rows 8..15.
      float s = d[0] + d[1] + d[2] + d[3] + d[4] + d[5] + d[6] + d[7];
      s += __shfl_xor(s, 16);  // combine halves -> full 32-lane sum in all lanes
      out3[q] = s;
    }

    float mt = -__builtin_inff(), mm = -__builtin_inff();
#pragma unroll
    for (int i = 0; i < THREADS / 32; ++i) {
      mt = fmaxf(mt, redm[0][tid + 32 * i]);
      mm = fmaxf(mm, redm[1][tid + 32 * i]);
    }
#pragma unroll
    for (int off = 16; off >= 1; off >>= 1) {
      mt = fmaxf(mt, __shfl_xor(mt, off));
      mm = fmaxf(mm, __shfl_xor(mm, off));
    }

    if (tid == 0) {
      float* p = partials + ((size_t)slice * SPLIT + part) * 5;
      p[0] = out3[0];  // sum target
      p[1] = out3[1];  // sum diff2
      p[2] = out3[2];  // sum diff2*target
      p[3] = mt;       // max target
      p[4] = mm;       // max max_positiones
    }
  }
}

// ---------------------------------------------------------------------------
// Kernel B: fold SPLIT partials per slice, apply loss formula + masking +
// per-image nonzero-count normalization + final mean. One block, 128 threads.
// ---------------------------------------------------------------------------
__global__ __launch_bounds__(128)
void mismatch_final_kernel(const float* __restrict__ partials,
                           float* __restrict__ out) {
  const int t = threadIdx.x;  // slice id: b = t/16, c = t%16
  const float* p = partials + (size_t)t * SPLIT * 5;

  float s_t = 0.0f, s_d2 = 0.0f, s_d2t = 0.0f;
  float mt = -__builtin_inff(), mm = -__builtin_inff();
#pragma unroll
  for (int i = 0; i < SPLIT; ++i) {
    s_t   += p[i * 5 + 0];
    s_d2  += p[i * 5 + 1];
    s_d2t += p[i * 5 + 2];
    mt = fmaxf(mt, p[i * 5 + 3]);
    mm = fmaxf(mm, p[i * 5 + 4]);
  }

  const float m1  = s_d2t;
  const float d1  = s_t;
  const float m2  = s_d2 - s_d2t;          // sum diff2*(1-t)
  const float d2v = (float)HW - s_t;       // sum (1-t)
  const float loss = ALPHA * m1 / (d1 + SMOOTH)
                   + (1.0f - ALPHA) * m2 / (d2v + SMOOTH);
  const bool active = !((mt == 0.0f) && (mm == 0.0f));
  const float lv = active ? loss : 0.0f;

  __shared__ float lvs[128];
  __shared__ float imgs[8];
  lvs[t] = lv;
  __syncthreads();

  if (t < 8) {  // one thread per image
    float s = 0.0f, cnt = 0.0f;
#pragma unroll
    for (int c = 0; c < 16; ++c) {
      const float v = lvs[t * 16 + c];
      s += v;
      cnt += (v != 0.0f) ? 1.0f : 0.0f;  // reference divides by #nonzero
    }
    imgs[t] = s / cnt;                   // 0/0 -> NaN, matching reference
  }
  __syncthreads();

  if (t == 0) {
    float s = 0.0f;
#pragma unroll
    for (int b = 0; b < 8; ++b) s += imgs[b];
    out[0] = s / 8.0f;
  }
}

// ---------------------------------------------------------------------------
extern "C" void kernel_launch(void* const* d_in, const int* in_sizes, int n_in,
                              void* d_out, int out_size, void* d_ws, size_t ws_size,
                              hipStream_t stream) {
  const float* net_out = (const float*)d_in[0];
  const float* target  = (const float*)d_in[1];
  const float* maxpos  = (const float*)d_in[2];
  float* partials = (float*)d_ws;  // 128*8*5 floats = 20480 bytes

  mismatch_partial_kernel<<<SLICES * SPLIT, THREADS, 0, stream>>>(
      net_out, target, maxpos, partials);
  mismatch_final_kernel<<<1, 128, 0, stream>>>(partials, (float*)d_out);
}